// ScanCompensation_1297080123627
// MI455X (gfx1250) — compile-verified
//
#include <hip/hip_runtime.h>
#include <hip/hip_bf16.h>

#define HH 720
#define WW 1280
#define HW (HH * WW)            // 921600 = 256 * 3600
#define NUM_BINS 64
#define BIN_WIDTH_F 100000.0f

typedef __attribute__((ext_vector_type(2))) float v2f;
typedef __attribute__((ext_vector_type(4))) float v4f;
typedef __attribute__((ext_vector_type(8))) float v8f;

// ---- ordered-uint encoding so unsigned atomicMin == float min ----
__device__ __forceinline__ unsigned f2key(float f) {
    unsigned u = __float_as_uint(f);
    return (u & 0x80000000u) ? ~u : (u | 0x80000000u);
}
__device__ __forceinline__ float key2f(unsigned k) {
    unsigned u = (k & 0x80000000u) ? (k ^ 0x80000000u) : ~k;
    return __uint_as_float(u);
}

// WMMA column-sum step: A = ones(16x4) -> D[m][n] = sum_k B[k][n] + C[m][n]
__device__ __forceinline__ v8f wsum(v2f b, v8f c) {
    v2f ones = {1.0f, 1.0f};
    return __builtin_amdgcn_wmma_f32_16x16x4_f32(false, ones, false, b,
                                                 (short)0, c, false, false);
}

// ---------------- Phase 1: min of warped timestamps ----------------
__global__ void warp_min_kernel(const float* __restrict__ params,
                                const float* __restrict__ xs,
                                const float* __restrict__ ys,
                                const float* __restrict__ ts,
                                unsigned* __restrict__ minKey, int n) {
    const float a0 = params[0];
    const float a1 = params[1];
    float m = 3.4028235e38f;
    const int stride = gridDim.x * blockDim.x;
    const int tid = blockIdx.x * blockDim.x + threadIdx.x;
    const int n4 = n >> 2;
    const v4f* x4 = (const v4f*)xs;
    const v4f* y4 = (const v4f*)ys;
    const v4f* t4 = (const v4f*)ts;
    for (int i = tid; i < n4; i += stride) {
        v4f x = x4[i];
        v4f y = y4[i];
        v4f t = t4[i];
#pragma unroll
        for (int j = 0; j < 4; ++j)
            m = fminf(m, t[j] - a0 * x[j] - a1 * y[j]);
    }
    for (int i = (n4 << 2) + tid; i < n; i += stride)   // tail
        m = fminf(m, ts[i] - a0 * xs[i] - a1 * ys[i]);
    for (int off = 16; off > 0; off >>= 1)              // wave32 butterfly
        m = fminf(m, __shfl_xor(m, off, 32));
    if ((threadIdx.x & 31) == 0)
        atomicMin(minKey, f2key(m));
}

// ---------------- Phase 2: bilinear temporal scatter-add ----------------
__device__ __forceinline__ void scatter_one(float x, float y, float t, float p,
                                            float a0, float a1, float t_start,
                                            float* __restrict__ tensor) {
    float tw = t - a0 * x - a1 * y;
    float tn = (tw - t_start) / BIN_WIDTH_F;
    float t0f = floorf(tn);
    float wt = tn - t0f;
    int t0 = (int)t0f;
    int t1 = t0 + 1;
    t0 = min(max(t0, 0), NUM_BINS - 1);
    t1 = min(max(t1, 0), NUM_BINS - 1);
    int xi = (int)x;
    int yi = (int)y;
    if (xi >= 0 && xi < WW && yi >= 0 && yi < HH) {
        int sp = yi * WW + xi;
        unsafeAtomicAdd(tensor + (size_t)t0 * HW + sp, (1.0f - wt) * p);
        unsafeAtomicAdd(tensor + (size_t)t1 * HW + sp, wt * p);
    }
}

__global__ void scatter_kernel(const float* __restrict__ params,
                               const float* __restrict__ xs,
                               const float* __restrict__ ys,
                               const float* __restrict__ ts,
                               const float* __restrict__ pol,
                               const unsigned* __restrict__ minKey,
                               float* __restrict__ tensor, int n) {
    const float a0 = params[0];
    const float a1 = params[1];
    const float t_start = key2f(*minKey);
    const int stride = gridDim.x * blockDim.x;
    const int tid = blockIdx.x * blockDim.x + threadIdx.x;
    const int n4 = n >> 2;
    const v4f* x4 = (const v4f*)xs;
    const v4f* y4 = (const v4f*)ys;
    const v4f* t4 = (const v4f*)ts;
    const v4f* p4 = (const v4f*)pol;
    for (int i = tid; i < n4; i += stride) {
        v4f x = x4[i];
        v4f y = y4[i];
        v4f t = t4[i];
        v4f p = p4[i];
#pragma unroll
        for (int j = 0; j < 4; ++j)
            scatter_one(x[j], y[j], t[j], p[j], a0, a1, t_start, tensor);
    }
    for (int i = (n4 << 2) + tid; i < n; i += stride)   // tail
        scatter_one(xs[i], ys[i], ts[i], pol[i], a0, a1, t_start, tensor);
}

// ---------------- Phase 3: per-bin sum / sumsq via WMMA reduction ----------------
// Each wave streams 256 floats/iter via two b128 loads; 8 wmmas feed 4
// independent accumulator chains.  Wave total = (sum over 32 lanes of c[0])/2
// since D rows 0 and 8 (both = column sums) land duplicated across the wave.
__global__ void var_partial_kernel(const float* __restrict__ tensor,
                                   float* __restrict__ binSum,
                                   float* __restrict__ binSumSq) {
    const int bin = blockIdx.y;
    const float* base = tensor + (size_t)bin * HW;
    const int lane = threadIdx.x & 31;
    const int waveId = threadIdx.x >> 5;
    const int wavesPerBlock = blockDim.x >> 5;
    const int waveGlobal = blockIdx.x * wavesPerBlock + waveId;
    const int waveStride = gridDim.x * wavesPerBlock;
    const int nChunks = HW / 256;  // 3600, exact

    v8f cs0 = {}, cs1 = {}, cq0 = {}, cq1 = {};
    for (int chunk = waveGlobal; chunk < nChunks; chunk += waveStride) {
        const float* p = base + chunk * 256 + lane * 4;
        v4f b0 = *(const v4f*)p;          // floats [4*lane .. 4*lane+3]
        v4f b1 = *(const v4f*)(p + 128);  // second 128-float half
        v2f b0l = {b0.x, b0.y}, b0h = {b0.z, b0.w};
        v2f b1l = {b1.x, b1.y}, b1h = {b1.z, b1.w};
        cs0 = wsum(b0l, cs0);
        cs1 = wsum(b0h, cs1);
        cs0 = wsum(b1l, cs0);
        cs1 = wsum(b1h, cs1);
        cq0 = wsum(b0l * b0l, cq0);
        cq1 = wsum(b0h * b0h, cq1);
        cq0 = wsum(b1l * b1l, cq0);
        cq1 = wsum(b1h * b1h, cq1);
    }
    float s = cs0[0] + cs1[0];
    float q = cq0[0] + cq1[0];
    for (int off = 16; off > 0; off >>= 1) {
        s += __shfl_xor(s, off, 32);
        q += __shfl_xor(q, off, 32);
    }
    s *= 0.5f;  // rows 0 and 8 identical -> whole-wave sum double-counts
    q *= 0.5f;
    if (lane == 0) {
        unsafeAtomicAdd(&binSum[bin], s);
        unsafeAtomicAdd(&binSumSq[bin], q);
    }
}

// ---------------- Phase 4: variances -> mean loss ----------------
__global__ void finalize_kernel(const float* __restrict__ binSum,
                                const float* __restrict__ binSumSq,
                                float* __restrict__ loss) {
    __shared__ float vars[NUM_BINS];
    int b = threadIdx.x;
    const float n = (float)HW;
    float s = binSum[b];
    float q = binSumSq[b];
    vars[b] = (q - s * s / n) / (n - 1.0f);  // unbiased (ddof=1)
    __syncthreads();
    if (b == 0) {
        float acc = 0.0f;
        for (int i = 0; i < NUM_BINS; ++i) acc += vars[i];
        *loss = acc / (float)NUM_BINS;
    }
}

extern "C" void kernel_launch(void* const* d_in, const int* in_sizes, int n_in,
                              void* d_out, int out_size, void* d_ws, size_t ws_size,
                              hipStream_t stream) {
    const float* params = (const float*)d_in[0];
    const float* xs     = (const float*)d_in[1];
    const float* ys     = (const float*)d_in[2];
    const float* ts     = (const float*)d_in[3];
    const float* pol    = (const float*)d_in[4];
    // d_in[5] = num_bins scalar, statically NUM_BINS
    const int n = in_sizes[1];

    unsigned* minKey = (unsigned*)d_ws;
    float* binSum    = (float*)((char*)d_ws + 64);
    float* binSumSq  = binSum + NUM_BINS;
    float* tensor    = (float*)d_out;
    float* loss      = tensor + (size_t)NUM_BINS * HW;

    // per-call init (harness poisons but never re-zeros)
    hipMemsetAsync(minKey, 0xFF, sizeof(unsigned), stream);          // +inf key
    hipMemsetAsync(binSum, 0, 2 * NUM_BINS * sizeof(float), stream);
    hipMemsetAsync(tensor, 0, (size_t)NUM_BINS * HW * sizeof(float), stream);

    warp_min_kernel<<<2048, 256, 0, stream>>>(params, xs, ys, ts, minKey, n);
    scatter_kernel<<<4096, 256, 0, stream>>>(params, xs, ys, ts, pol, minKey,
                                             tensor, n);
    dim3 vgrid(32, NUM_BINS);
    var_partial_kernel<<<vgrid, 256, 0, stream>>>(tensor, binSum, binSumSq);
    finalize_kernel<<<1, NUM_BINS, 0, stream>>>(binSum, binSumSq, loss);
}